// PERIS_35648228557166
// MI455X (gfx1250) — compile-verified
//
#include <hip/hip_runtime.h>
#include <hip/hip_bf16.h>

#define B_   1024
#define L_   512
#define T_   128
#define K_   64
#define THR_ 500000
#define TAU_ 0.1f

typedef __attribute__((ext_vector_type(2))) float v2f;
typedef __attribute__((ext_vector_type(8))) float v8f;

// One block per (batch element, pipeline). pipeline 0: item/history/time_bins,
// pipeline 1: user/neighbors/neighbors_freqbin.
__global__ __launch_bounds__(256) void peris_sim_hist_kernel(
    const int* __restrict__ users, const int* __restrict__ items,
    const int* __restrict__ times,
    const int* __restrict__ history, const int* __restrict__ pack_times,
    const float* __restrict__ time_bins,
    const int* __restrict__ neighbors, const int* __restrict__ neighbors_time,
    const float* __restrict__ neighbors_freqbin,
    const float* __restrict__ ebd_user, const float* __restrict__ ebd_item,
    float* __restrict__ out_virt, float* __restrict__ out_feat_u,
    float* __restrict__ out_nlab, float* __restrict__ out_feat_n)
{
    __shared__ __align__(16) float qs[K_];   // query embedding
    __shared__ float wgt[L_];                // per-l masked similarity weights
    __shared__ float s_lab;                  // neighbor label accumulator

    const int b    = blockIdx.x;
    const int pipe = blockIdx.y;             // uniform per block
    const int tid  = threadIdx.x;

    const int*   gidx  = pipe ? neighbors         : history;
    const int*   gtime = pipe ? neighbors_time    : pack_times;
    const float* tab   = pipe ? ebd_user          : ebd_item;
    const float* bins  = pipe ? neighbors_freqbin : time_bins;
    float*     outfeat = pipe ? out_feat_n        : out_feat_u;
    const int    qidx  = pipe ? users[b]          : items[b];

    if (tid == 0) s_lab = 0.0f;
    if (tid < 16) {
        ((float4*)qs)[tid] = ((const float4*)(tab + (size_t)qidx * K_))[tid];
    }
    __syncthreads();

    // |q|^2 (redundant per thread; LDS broadcast reads are cheap)
    float qn2 = 0.0f;
    #pragma unroll
    for (int k4 = 0; k4 < K_ / 4; ++k4) {
        float4 q = ((const float4*)qs)[k4];
        qn2 += q.x * q.x + q.y * q.y + q.z * q.z + q.w * q.w;
    }

    // ---- Phase 1: similarities + masks -> wgt[0..511] in LDS ----
    float labacc = 0.0f;
    #pragma unroll
    for (int rep = 0; rep < 2; ++rep) {
        const int l   = tid + rep * 256;
        const int idx = gidx [(size_t)b * L_ + l];
        const int tt  = gtime[(size_t)b * L_ + l];
        const float4* e4 = (const float4*)(tab + (size_t)idx * K_);
        float dot = 0.0f, en2 = 0.0f;
        #pragma unroll
        for (int k4 = 0; k4 < K_ / 4; ++k4) {
            float4 e = e4[k4];
            float4 q = ((const float4*)qs)[k4];
            dot += q.x * e.x + q.y * e.y + q.z * e.z + q.w * e.w;
            en2 += e.x * e.x + e.y * e.y + e.z * e.z + e.w * e.w;
        }
        const float cosv = dot * rsqrtf(qn2 * en2);
        const float sim  = fminf((cosv + 1.0f) * 0.5f + TAU_, 1.0f);
        float wl;
        if (pipe) {
            const bool mask_n = (idx != 0);
            wl      = (mask_n && (tt <  THR_)) ? sim : 0.0f;
            labacc += (mask_n && (tt >= THR_)) ? sim : 0.0f;
        } else {
            wl = ((tt != 0) && (tt < THR_)) ? sim : 0.0f;
        }
        wgt[l] = wl;
    }
    if (pipe) atomicAdd(&s_lab, labacc);   // ds_add_f32
    __syncthreads();

    if (tid == 0) {
        if (pipe) out_nlab[b] = (s_lab > 1.0f) ? 1.0f : 0.0f;
        else      out_virt[b] = (times[b] >= THR_) ? 1.0f : 0.0f;
    }

    // ---- Phase 2: feat[t] = sum_l wgt[l] * bins[b,l,t] via f32 WMMA ----
    // Wave w owns t-tile [16w, 16w+16); K-step over l in chunks of 4.
    const int wave = tid >> 5;
    const int lane = tid & 31;
    const int grp  = lane >> 4;      // half-wave: K rows {0,1} vs {2,3}
    const int tcol = lane & 15;
    const int t0   = wave * 16;
    const float* binsb = bins + (size_t)b * L_ * T_ + t0 + tcol;

    v8f c = {};
    for (int l0 = 0; l0 < L_; l0 += 4) {
        const int row   = l0 + grp * 2;
        const float* p  = binsb + (size_t)row * T_;
        __builtin_prefetch(p + 4 * T_, 0, 1);          // global_prefetch_b8
        v2f av; av.x = wgt[row]; av.y = wgt[row + 1];  // A: w broadcast to 16 rows
        v2f bv; bv.x = p[0];     bv.y = p[T_];         // B: bins[4 x 16] tile
        c = __builtin_amdgcn_wmma_f32_16x16x4_f32(false, av, false, bv,
                                                  (short)0, c, false, false);
    }
    // D row 0 (lanes 0..15, VGPR 0) holds feat[t0..t0+15]
    if (lane < 16) outfeat[(size_t)b * T_ + t0 + lane] = c[0];
}

extern "C" void kernel_launch(void* const* d_in, const int* in_sizes, int n_in,
                              void* d_out, int out_size, void* d_ws, size_t ws_size,
                              hipStream_t stream) {
    const int*   users            = (const int*)  d_in[0];
    const int*   items            = (const int*)  d_in[1];
    const int*   times            = (const int*)  d_in[2];
    const int*   history          = (const int*)  d_in[3];
    const int*   pack_times       = (const int*)  d_in[4];
    const float* time_bins        = (const float*)d_in[5];
    const int*   neighbors        = (const int*)  d_in[6];
    const int*   neighbors_time   = (const int*)  d_in[7];
    const float* neighbors_freqbin= (const float*)d_in[8];
    const float* ebd_user         = (const float*)d_in[9];
    const float* ebd_item         = (const float*)d_in[10];

    float* out        = (float*)d_out;
    float* out_virt   = out;                       // [B]
    float* out_feat_u = out + B_;                  // [B,T]
    float* out_nlab   = out + B_ + B_ * T_;        // [B]
    float* out_feat_n = out + B_ + B_ * T_ + B_;   // [B,T]

    dim3 grid(B_, 2);
    peris_sim_hist_kernel<<<grid, 256, 0, stream>>>(
        users, items, times, history, pack_times, time_bins,
        neighbors, neighbors_time, neighbors_freqbin, ebd_user, ebd_item,
        out_virt, out_feat_u, out_nlab, out_feat_n);
}